// CombinedBoundaryLoss_50981261804140
// MI455X (gfx1250) — compile-verified
//
#include <hip/hip_runtime.h>
#include <math.h>

// CombinedBoundaryLoss for MI455X (gfx1250), wave32.
// B=4, C=1, H=W=96, N=9216. Output: single f32 scalar.

#define BATCH 4
#define HH 96
#define WW 96
#define NPIX (HH * WW)                 // 9216
#define ROWS_PER_BLOCK 128             // 8 waves * 16 rows
#define BLOCKS_PER_BATCH (NPIX / ROWS_PER_BLOCK)  // 72
#define HD_WS_BASE 32                  // ws[32 ..] = per-block Hausdorff partials

typedef __attribute__((ext_vector_type(2))) float v2f;
typedef __attribute__((ext_vector_type(8))) float v8f;

// ---------------------------------------------------------------------------
// Kernel 1: edge detection + dice sums + focal sum. One block per batch item,
// deterministic LDS tree reduction (no FP atomics).
// ws layout: [0..3]=inter, [4..7]=psum, [8..11]=tsum, [12..15]=inter_e,
//            [16..19]=te_sum, [20..23]=focal partial (per batch)
// ---------------------------------------------------------------------------
__global__ void __launch_bounds__(256) stats_kernel(const float* __restrict__ pred,
                                                    const float* __restrict__ target,
                                                    float* __restrict__ ws) {
  const int b = blockIdx.x;
  const int tid = threadIdx.x;
  const float* p = pred + b * NPIX;
  const float* t = target + b * NPIX;

  float inter = 0.f, psum = 0.f, tsum = 0.f, inter_e = 0.f, te = 0.f, foc = 0.f;

  for (int n = tid; n < NPIX; n += 256) {
    int y = n / WW;
    int x = n - y * WW;
    float tc = t[n];
    float tu = (y > 0)      ? t[n - WW] : 0.f;
    float td = (y < HH - 1) ? t[n + WW] : 0.f;
    float tl = (x > 0)      ? t[n - 1]  : 0.f;
    float tr = (x < WW - 1) ? t[n + 1]  : 0.f;
    // 3x3 Laplacian, zero (SAME) padding; target is {0,1} so this is exact.
    float edge = tu + td + tl + tr - 4.f * tc;
    float em = (fabsf(edge) > 0.f) ? 1.f : 0.f;

    float pv = p[n];
    float prob = 1.f / (1.f + expf(-pv));

    inter   += prob * tc;
    psum    += prob;
    tsum    += tc;
    inter_e += prob * em;
    te      += em;

    // stable BCE-with-logits
    float ce = fmaxf(pv, 0.f) - pv * tc + log1pf(expf(-fabsf(pv)));
    float pt = prob * tc + (1.f - prob) * (1.f - tc);
    float omp = 1.f - pt;
    float focal = 0.25f * omp * omp * ce;      // F_ALPHA=0.25, F_GAMMA=2
    foc += (1.f + 3.f * em) * focal;           // F_BW=3
  }

  __shared__ float red[256];
  float vals[6] = {inter, psum, tsum, inter_e, te, foc};
  for (int k = 0; k < 6; ++k) {
    red[tid] = vals[k];
    __syncthreads();
    for (int s = 128; s > 0; s >>= 1) {
      if (tid < s) red[tid] += red[tid + s];
      __syncthreads();
    }
    if (tid == 0) ws[k * 4 + b] = red[0];
    __syncthreads();
  }
}

// ---------------------------------------------------------------------------
// Kernel 2: SoftHausdorff, flash-attention style over the 9216x9216 squared-
// distance matrix. dot(i,j) = x_i*x_j + y_i*y_j via V_WMMA_F32_16X16X4_F32
// (K=4, slots 2..3 zero-padded); d2 = sq_i + sq_j - 2*dot (exact in f32,
// coords are small integers). Online masked softmax per row with per-column-
// lane partial state, merged with 4 shfl_xor steps (width 16 = one half-wave,
// matching the 16x16 f32 C/D lane layout).
//
// Grid: BATCH*72 blocks x 256 threads (8 waves, 16 rows each).
// Each block writes one f32 partial sum_i pred_i*min_dists_i to ws[32+blk].
// ---------------------------------------------------------------------------
__global__ void __launch_bounds__(256) hd_kernel(const float* __restrict__ pred,
                                                 const float* __restrict__ target,
                                                 float* __restrict__ ws) {
  const int blk  = blockIdx.x;
  const int b    = blk / BLOCKS_PER_BATCH;
  const int rblk = blk % BLOCKS_PER_BATCH;
  const int tid  = threadIdx.x;
  const int lane = tid & 31;
  const int wv   = tid >> 5;           // wave id 0..7
  const int half = lane >> 4;          // 0: rows r0..r0+7, 1: rows r0+8..r0+15
  const int c    = lane & 15;          // column slot within tile (C/D layout: N = lane%16)

  // Target mask for this batch item lives in LDS (36 KB of 320 KB/WGP).
  __shared__ float lds_t[NPIX];
  __shared__ float wpart[16];
  for (int n = tid; n < NPIX; n += 256) lds_t[n] = target[b * NPIX + n];
  __syncthreads();

  const int r0      = rblk * ROWS_PER_BLOCK + wv * 16;
  const int rowbase = r0 + half * 8;   // the 8 rows this lane's D slots hold

  // A fragment (16x4 f32, 2 VGPRs): lanes 0-15 -> (M=lane, K=0/1) = (x_r, y_r),
  // lanes 16-31 -> K=2/3 zero padding.
  v2f afrag;
  {
    int r  = r0 + c;
    int ry = r / WW;
    int rx = r - ry * WW;
    afrag[0] = half ? 0.f : (float)rx;
    afrag[1] = half ? 0.f : (float)ry;
  }

  float sqrow[8];
#pragma unroll
  for (int v = 0; v < 8; ++v) {
    int r  = rowbase + v;
    int ry = r / WW;
    int rx = r - ry * WW;
    sqrow[v] = (float)(rx * rx + ry * ry);
  }

  // Online softmax state: per lane, per row v over this lane's column stream.
  float m[8], den[8], num[8];
#pragma unroll
  for (int v = 0; v < 8; ++v) { m[v] = -1e30f; den[v] = 0.f; num[v] = 0.f; }

  v8f czero = {};

  for (int j0 = 0; j0 < NPIX; j0 += 16) {
    int j  = j0 + c;
    int jy = j / WW;
    int jx = j - jy * WW;
    float fx = (float)jx, fy = (float)jy;

    // B fragment (4x16 f32): lanes 0-15 -> (K=0/1, N=lane) = (x_j, y_j),
    // lanes 16-31 -> K=2/3 zero padding.
    v2f bfrag;
    bfrag[0] = half ? 0.f : fx;
    bfrag[1] = half ? 0.f : fy;

    // dot[v] = x_row * x_col + y_row * y_col for (row = rowbase+v, col = j)
    v8f dot = __builtin_amdgcn_wmma_f32_16x16x4_f32(
        /*neg_a=*/false, afrag, /*neg_b=*/false, bfrag,
        /*c_mod=*/(short)0, czero, /*reuse_a=*/false, /*reuse_b=*/false);

    float sqj = fx * fx + fy * fy;
    float tm  = lds_t[j];
    if (tm > 0.5f) {                   // masked softmax column; reconverges before next WMMA
#pragma unroll
      for (int v = 0; v < 8; ++v) {
        float d2 = fmaf(-2.f, dot[v], sqrow[v] + sqj);
        d2 = fmaxf(d2, 0.f);
        float l  = -100.f * d2;        // -dists_sq / HD_TEMP
        float mn = fmaxf(m[v], l);
        float sc = __expf(m[v] - mn);  // m init -1e30 -> sc underflows to 0 safely
        float e  = __expf(l - mn);
        den[v] = den[v] * sc + e;
        num[v] = num[v] * sc + e * d2;
        m[v]   = mn;
      }
    }
  }

  // Merge the 16 per-column partial softmax states per row (split-softmax merge),
  // confined to each half-wave (width 16).
#pragma unroll
  for (int v = 0; v < 8; ++v) {
    for (int off = 1; off < 16; off <<= 1) {
      float m2 = __shfl_xor(m[v],   off, 16);
      float e2 = __shfl_xor(den[v], off, 16);
      float n2 = __shfl_xor(num[v], off, 16);
      float mn = fmaxf(m[v], m2);
      float s1 = __expf(m[v] - mn);
      float s2 = __expf(m2  - mn);
      den[v] = den[v] * s1 + e2 * s2;
      num[v] = num[v] * s1 + n2 * s2;
      m[v]   = mn;
    }
  }

  // loss_p partial: sum_v pred[row] * (num/den). All 16 lanes of a half are
  // identical after the merge -> one representative lane per half contributes.
  float lp = 0.f;
  const float* pp = pred + b * NPIX;
#pragma unroll
  for (int v = 0; v < 8; ++v) {
    float md = (den[v] > 0.f) ? (num[v] / den[v]) : 0.f;
    lp += pp[rowbase + v] * md;
  }
  if (c == 0) wpart[wv * 2 + half] = lp;
  __syncthreads();
  if (tid == 0) {
    float s = 0.f;
    for (int k = 0; k < 16; ++k) s += wpart[k];   // fixed order: deterministic
    ws[HD_WS_BASE + blk] = s;
  }
}

// ---------------------------------------------------------------------------
// Kernel 3: fold everything into the scalar loss (fixed summation order).
// ---------------------------------------------------------------------------
__global__ void finalize_kernel(const float* __restrict__ ws, float* __restrict__ out) {
  if (threadIdx.x != 0 || blockIdx.x != 0) return;
  const float S = 1e-5f;               // D_SMOOTH
  float dice_all_sum = 0.f, dice_e_sum = 0.f, te_total = 0.f, focal_sum = 0.f;
  float hd = 0.f;
  for (int b = 0; b < BATCH; ++b) {
    float inter   = ws[0 * 4 + b];
    float psum    = ws[1 * 4 + b];
    float tsum    = ws[2 * 4 + b];
    float inter_e = ws[3 * 4 + b];
    float te      = ws[4 * 4 + b];
    focal_sum += ws[5 * 4 + b];
    te_total  += te;
    dice_all_sum += (2.f * inter + S) / (psum + tsum + S);
    dice_e_sum   += (2.f * inter_e + S) / (inter_e + te + S);  // faithful to reference

    float lp = 0.f;
    for (int k = 0; k < BLOCKS_PER_BATCH; ++k)
      lp += ws[HD_WS_BASE + b * BLOCKS_PER_BATCH + k];
    // loss_t == 0 exactly (col-min of dists_sq contains the zero diagonal).
    hd += (tsum > 0.f) ? lp : 0.f;     // jnp.where(m.any(), ...)
  }
  float loss_all   = 1.f - dice_all_sum / (float)BATCH;
  float loss_edge  = (te_total > 0.f) ? (1.f - dice_e_sum / (float)BATCH) : 0.f;
  float dice_loss  = loss_all + 2.0f * loss_edge;               // D_BW=2
  float focal_loss = focal_sum / (float)(BATCH * NPIX);
  float hd_loss    = hd / (float)BATCH;
  out[0] = 1.0f * dice_loss + 0.5f * focal_loss + 0.1f * hd_loss;
}

// ---------------------------------------------------------------------------
extern "C" void kernel_launch(void* const* d_in, const int* in_sizes, int n_in,
                              void* d_out, int out_size, void* d_ws, size_t ws_size,
                              hipStream_t stream) {
  (void)in_sizes; (void)n_in; (void)out_size; (void)ws_size;
  const float* pred   = (const float*)d_in[0];
  const float* target = (const float*)d_in[1];
  float* ws  = (float*)d_ws;
  float* out = (float*)d_out;

  hipLaunchKernelGGL(stats_kernel, dim3(BATCH), dim3(256), 0, stream, pred, target, ws);
  hipLaunchKernelGGL(hd_kernel, dim3(BATCH * BLOCKS_PER_BATCH), dim3(256), 0, stream,
                     pred, target, ws);
  hipLaunchKernelGGL(finalize_kernel, dim3(1), dim3(32), 0, stream, ws, out);
}